// SoftNMS_3264175145228
// MI455X (gfx1250) — compile-verified
//
#include <hip/hip_runtime.h>
#include <stdint.h>

// Problem constants (match reference)
#define NBOX 8192
#define NLBL 8
#define TPB  256          // 8 waves of 32 on CDNA5
#define MAXP 2048         // LDS-resident path capacity per label
#define SCORE_THR 0.001f
#define INV_SIGMA 2.0f    // 1/SIGMA, SIGMA=0.5

// ---------------------------------------------------------------------------
// Shared-memory layout for the per-label soft-NMS block.
// NOTE: single __shared__ object per kernel -> group segment offset 0, so
// __builtin_offsetof(SM, field) is the LDS byte address used by the async
// global->LDS instructions.
// ---------------------------------------------------------------------------
struct SM {
    float box[MAXP][4];     // x1,y1,x2,y2 (AoS, filled by async b128 gather)
    float sc[MAXP];         // mutable scores (filled by async b32 gather)
    float ar[MAXP];         // areas
    unsigned char al[MAXP]; // alive flags
    float rs[8];            // per-wave argmax score
    int   rp[8];            // per-wave argmax local index
    int   bp;               // block argmax local index
    int   cnt;              // alive-rank accumulator (ds_add target)
};

// ---------------------------------------------------------------------------
// K1: per-label stable compaction of global indices (wave32 ballot).
// order[l*NBOX + p] = global index of p-th box of label l (ascending),
// which equals jnp.argsort(where(lm, ar, ar+n))[:n_l].
// ---------------------------------------------------------------------------
__global__ void k_partition(const int* __restrict__ labels,
                            int* __restrict__ order,
                            int* __restrict__ cnt) {
    const int l = blockIdx.x;          // one wave (32 threads) per label
    const int lane = threadIdx.x;
    int base = 0;
    for (int start = 0; start < NBOX; start += 32) {
        const int i = start + lane;    // NBOX is a multiple of 32
        const bool m = (labels[i] == l);
        const unsigned msk = __builtin_amdgcn_ballot_w32(m);
        if (m)
            order[l * NBOX + base + __popc(msk & ((1u << lane) - 1u))] = i;
        base += __popc(msk);
    }
    if (lane == 0) cnt[l] = base;
}

// ---------------------------------------------------------------------------
// Core serial soft-NMS loop (pointers may target LDS or global scratch).
// Faithful to the reference:
//   i   = argmax over alive scores (first occurrence on ties)
//   pos = # alive with index < i   (rank in compacted working set)
//   kp[k] = pos;  decay = exp(-iou^2/sigma) on alive others;
//   alive = others & (score >= thr)
// ---------------------------------------------------------------------------
__device__ void softnms_core(SM& sm, float* bx, float* sc, float* ar,
                             unsigned char* al, int n,
                             int* __restrict__ kp, int* __restrict__ kc_out) {
    const int tid  = threadIdx.x;
    const int lane = tid & 31;
    const int wid  = tid >> 5;
    int kc = 0;

    for (int iter = 0; iter < n; ++iter) {
        if (tid == 0) { sm.bp = 0x7fffffff; sm.cnt = 0; }
        __syncthreads();                       // prior decay writes visible

        // ---- block argmax over alive scores ----
        float bs = -3.4e38f; int bp = 0x7fffffff;
        for (int p = tid; p < n; p += TPB) {
            if (al[p]) {
                const float s = sc[p];
                if (s > bs || (s == bs && p < bp)) { bs = s; bp = p; }
            }
        }
        for (int o = 16; o > 0; o >>= 1) {
            const float os = __shfl_down(bs, o);
            const int   op = __shfl_down(bp, o);
            if (os > bs || (os == bs && op < bp)) { bs = os; bp = op; }
        }
        if (lane == 0) { sm.rs[wid] = bs; sm.rp[wid] = bp; }
        __syncthreads();
        if (wid == 0) {
            bs = (lane < 8) ? sm.rs[lane] : -3.4e38f;
            bp = (lane < 8) ? sm.rp[lane] : 0x7fffffff;
            for (int o = 4; o > 0; o >>= 1) {
                const float os = __shfl_down(bs, o);
                const int   op = __shfl_down(bp, o);
                if (os > bs || (os == bs && op < bp)) { bs = os; bp = op; }
            }
            if (lane == 0) sm.bp = bp;
        }
        __syncthreads();
        const int q = sm.bp;
        if (q == 0x7fffffff) break;            // no alive boxes left

        // ---- pos = # alive strictly below q (ds_add reduction) ----
        int c = 0;
        for (int p = tid; p < q; p += TPB) c += al[p] ? 1 : 0;
        for (int o = 16; o > 0; o >>= 1) c += __shfl_down(c, o);
        if (lane == 0) atomicAdd(&sm.cnt, c);
        __syncthreads();
        if (tid == 0) kp[kc] = sm.cnt;         // the reference's quirk: rank, not q
        kc++;

        // ---- gaussian decay of alive others ----
        const float qx1 = bx[4*q+0], qy1 = bx[4*q+1];
        const float qx2 = bx[4*q+2], qy2 = bx[4*q+3];
        const float qa  = ar[q];
        for (int p = tid; p < n; p += TPB) {
            if (p == q) { al[p] = 0; continue; }   // remove selected box
            if (!al[p]) continue;
            const float x1 = fmaxf(qx1, bx[4*p+0]);
            const float y1 = fmaxf(qy1, bx[4*p+1]);
            const float x2 = fminf(qx2, bx[4*p+2]);
            const float y2 = fminf(qy2, bx[4*p+3]);
            const float w  = fmaxf(x2 - x1, 0.0f);
            const float h  = fmaxf(y2 - y1, 0.0f);
            const float inter = w * h;
            const float iou = inter / (qa + ar[p] - inter + 1e-8f);
            const float s = sc[p] * expf(-(iou * iou) * INV_SIGMA);
            sc[p] = s;
            if (!(s >= SCORE_THR)) al[p] = 0;
        }
        __syncthreads();
    }
    if (tid == 0) *kc_out = kc;
}

// ---------------------------------------------------------------------------
// K2: one block per label. Stage gathered boxes/scores into LDS with the
// gfx1250 async global->LDS path (per-lane VGPR global address, per-lane LDS
// destination; tracked with ASYNCcnt), then run the serial loop fully in LDS.
// ---------------------------------------------------------------------------
__global__ void k_softnms(const float4* __restrict__ boxes,
                          const float*  __restrict__ scores,
                          const int* __restrict__ order,
                          const int* __restrict__ cnt,
                          float* __restrict__ gbox, float* __restrict__ gsc,
                          float* __restrict__ gar, unsigned char* __restrict__ gal,
                          int* __restrict__ kp, int* __restrict__ kc) {
    __shared__ SM sm;
    const int l   = blockIdx.x;
    const int n   = cnt[l];
    const int tid = threadIdx.x;
    const int* ord = order + l * NBOX;

    if (n <= MAXP) {
        // async gather: boxes[g] (16B) and scores[g] (4B) -> LDS
        for (int p = tid; p < n; p += TPB) {
            const int g = ord[p];
            const unsigned long long gaB =
                (unsigned long long)(uintptr_t)(const void*)(boxes + g);
            const unsigned ldsB =
                (unsigned)(__builtin_offsetof(SM, box) + 16u * (unsigned)p);
            asm volatile("global_load_async_to_lds_b128 %0, %1, off"
                         :: "v"(ldsB), "v"(gaB) : "memory");
            const unsigned long long gaS =
                (unsigned long long)(uintptr_t)(const void*)(scores + g);
            const unsigned ldsS =
                (unsigned)(__builtin_offsetof(SM, sc) + 4u * (unsigned)p);
            asm volatile("global_load_async_to_lds_b32 %0, %1, off"
                         :: "v"(ldsS), "v"(gaS) : "memory");
        }
        asm volatile("s_wait_asynccnt 0" ::: "memory");
        __syncthreads();
        for (int p = tid; p < n; p += TPB) {
            sm.ar[p] = (sm.box[p][2] - sm.box[p][0]) * (sm.box[p][3] - sm.box[p][1]);
            sm.al[p] = 1;
        }
        __syncthreads();
        softnms_core(sm, &sm.box[0][0], sm.sc, sm.ar, sm.al, n,
                     kp + l * NBOX, kc + l);
    } else {
        // fallback: working set in global scratch (still one block per label)
        float* bx = gbox + (size_t)l * NBOX * 4;
        float* sc = gsc + l * NBOX;
        float* ar = gar + l * NBOX;
        unsigned char* al = gal + (size_t)l * NBOX;
        for (int p = tid; p < n; p += TPB) {
            const int g = ord[p];
            const float4 b = boxes[g];
            bx[4*p+0] = b.x; bx[4*p+1] = b.y; bx[4*p+2] = b.z; bx[4*p+3] = b.w;
            sc[p] = scores[g];
            ar[p] = (b.z - b.x) * (b.w - b.y);
            al[p] = 1;
        }
        __threadfence_block();
        __syncthreads();
        softnms_core(sm, bx, sc, ar, al, n, kp + l * NBOX, kc + l);
    }
}

// K3: tiny exclusive prefix over 8 per-label keep counts.
__global__ void k_offsets(const int* __restrict__ kc, int* __restrict__ off) {
    if (threadIdx.x == 0) {
        int a = 0;
        for (int l = 0; l < NLBL; ++l) { off[l] = a; a += kc[l]; }
        off[NLBL] = a;
    }
}

// K4: zero the packed output (reference zero-pads past `count`).
__global__ void k_zero(float* __restrict__ out, int n) {
    const int i = blockIdx.x * blockDim.x + threadIdx.x;
    if (i < n) out[i] = 0.0f;
}

// K5: gather kept boxes into the packed output; write scalar count.
__global__ void k_emit(const float4* __restrict__ boxes,
                       const float*  __restrict__ scores,
                       const int* __restrict__ labels,
                       const int* __restrict__ order,
                       const int* __restrict__ kp,
                       const int* __restrict__ kc,
                       const int* __restrict__ off,
                       float* __restrict__ out) {
    const int l = blockIdx.x;
    const int nk = kc[l];
    const int o  = off[l];
    for (int k = threadIdx.x; k < nk; k += TPB) {
        const int pos = kp[l * NBOX + k];       // rank recorded by softnms
        const int g   = order[l * NBOX + pos];  // order[kp] (reference quirk)
        const float4 b = boxes[g];
        const int row = o + k;
        out[4*row+0] = b.x; out[4*row+1] = b.y;
        out[4*row+2] = b.z; out[4*row+3] = b.w;
        out[4*NBOX + row] = scores[g];
        out[5*NBOX + row] = (float)labels[g];
    }
    if (l == 0 && threadIdx.x == 0) out[6*NBOX] = (float)off[NLBL];
}

// ---------------------------------------------------------------------------
// Host-side launcher (graph-capture safe: only kernel launches on `stream`).
// Workspace layout (~2.2 MB): counts/offsets + order + kp + global fallback
// arrays for the n>MAXP path.
// ---------------------------------------------------------------------------
extern "C" void kernel_launch(void* const* d_in, const int* in_sizes, int n_in,
                              void* d_out, int out_size, void* d_ws, size_t ws_size,
                              hipStream_t stream) {
    const float4* boxes  = (const float4*)d_in[0];   // [N,4] f32
    const float*  scores = (const float*)d_in[1];    // [N]   f32
    const int*    labels = (const int*)d_in[2];      // [N]   i32
    float* out = (float*)d_out;                      // 4N + N + N + 1 floats

    // workspace carve-up
    int* w_cnt   = (int*)d_ws;                  // [L]
    int* w_kc    = w_cnt + NLBL;                // [L]
    int* w_off   = w_kc + NLBL;                 // [L+1]
    int* w_order = (int*)d_ws + 32;             // [L*N]
    int* w_kp    = w_order + NLBL * NBOX;       // [L*N]
    float* w_box = (float*)(w_kp + NLBL * NBOX);          // [L*N*4]
    float* w_sc  = w_box + 4 * NLBL * NBOX;               // [L*N]
    float* w_ar  = w_sc + NLBL * NBOX;                    // [L*N]
    unsigned char* w_al = (unsigned char*)(w_ar + NLBL * NBOX); // [L*N] bytes

    k_partition<<<NLBL, 32, 0, stream>>>(labels, w_order, w_cnt);
    k_softnms<<<NLBL, TPB, 0, stream>>>(boxes, scores, w_order, w_cnt,
                                        w_box, w_sc, w_ar, w_al, w_kp, w_kc);
    k_offsets<<<1, 32, 0, stream>>>(w_kc, w_off);
    k_zero<<<(out_size + TPB - 1) / TPB, TPB, 0, stream>>>(out, out_size);
    k_emit<<<NLBL, TPB, 0, stream>>>(boxes, scores, labels,
                                     w_order, w_kp, w_kc, w_off, out);
}